// Decoder_21371757265153
// MI455X (gfx1250) — compile-verified
//
#include <hip/hip_runtime.h>
#include <math.h>

typedef float v2f __attribute__((ext_vector_type(2)));
typedef float v8f __attribute__((ext_vector_type(8)));

#define RNN_H   384
#define HIDSZ   384
#define ENCW    768
#define POS_EMBW 128
#define LABELS  33
#define BATCH   16
#define CHAR_NUM 1024
#define KLIN    (RNN_H + ENCW)     // 1152
#define KCMB    (POS_EMBW + ENCW)  // 896
#define NTHREADS 768               // 24 wave32s

__device__ __forceinline__ float sigmoidf_(float x) { return 1.0f / (1.0f + expf(-x)); }
__device__ __forceinline__ v2f ld2(const float* p) { return *(const v2f*)p; }

// D = A(16x4,f32) * B(4x16,f32) + C(16x16,f32)
__device__ __forceinline__ v8f wmma4(v2f a, v2f b, v8f c) {
    return __builtin_amdgcn_wmma_f32_16x16x4_f32(
        /*neg_a=*/false, a, /*neg_b=*/false, b,
        /*c_mod=*/(short)0, c, /*reuse_a=*/false, /*reuse_b=*/false);
}

extern "C" __global__ __launch_bounds__(NTHREADS, 1)
void decoder_seq_kernel(const float* __restrict__ enc,        // [16,1024,768]
                        const int*   __restrict__ char_sizes, // [16]
                        const int*   __restrict__ label_is_sep,
                        const int*   __restrict__ label_pos_id,
                        const float* __restrict__ posEmb,     // [32,128]
                        const float* __restrict__ W_ih,       // [1536,384]
                        const float* __restrict__ W_hh,       // [1536,384]
                        const float* __restrict__ b_ih,       // [1536]
                        const float* __restrict__ b_hh,       // [1536]
                        const float* __restrict__ W_lin,      // [33,1152]
                        const float* __restrict__ b_lin,      // [33]
                        const float* __restrict__ W_cmb,      // [384,896]
                        const float* __restrict__ b_cmb,      // [384]
                        float* __restrict__ out)              // [16*1024,33]
{
    extern __shared__ float smem[];
    float* H    = smem;                      // 16*384
    float* C    = H   + BATCH * RNN_H;       // 16*384
    float* X    = C   + BATCH * RNN_H;       // 16*384
    float* E    = X   + BATCH * HIDSZ;       // 16*768  enc_t
    float* S    = E   + BATCH * ENCW;        // 16*768  prefix[t]
    float* W0   = S   + BATCH * ENCW;        // 16*768  prefix[word start]
    float* AVG  = W0  + BATCH * ENCW;        // 16*768
    float* LPE  = AVG + BATCH * ENCW;        // 16*128
    float* LG   = LPE + BATCH * POS_EMBW;    // 16*48 logits (padded)
    float* INVD = LG  + BATCH * 48;          // 16
    int*   ib   = (int*)(INVD + 16);
    int *wlen = ib, *nw = ib + 16, *npos = ib + 32, *plast = ib + 48;
    int *sepf = ib + 64, *validf = ib + 80, *dolstm = ib + 96;
    int *lpec = ib + 112, *plo = ib + 128;

    const int tid  = threadIdx.x;
    const int lane = tid & 31;
    const int wv   = tid >> 5;       // 0..23
    const int half = lane >> 4;      // 0/1
    const int l15  = lane & 15;
    const int koff = 2 * half;       // A/B K sub-offset per half-wave

    // ---------------- init: h0,c0 = lstm_cell(0,0,0); zero state ----------------
    if (tid < RNN_H) {
        int j = tid;
        float bi = b_ih[j]             + b_hh[j];
        float bg = b_ih[2*RNN_H + j]   + b_hh[2*RNN_H + j];
        float bo = b_ih[3*RNN_H + j]   + b_hh[3*RNN_H + j];
        float c0 = sigmoidf_(bi) * tanhf(bg);     // f-gate * c(=0) drops out
        float h0 = sigmoidf_(bo) * tanhf(c0);
        for (int m = 0; m < BATCH; ++m) { H[m*RNN_H + j] = h0; C[m*RNN_H + j] = c0; }
    }
    for (int i = tid; i < BATCH * POS_EMBW; i += NTHREADS) LPE[i] = 0.f;
    for (int m = 0; m < BATCH; ++m) { S[m*ENCW + tid] = 0.f; W0[m*ENCW + tid] = 0.f; }
    if (tid < BATCH) { wlen[tid] = 0; nw[tid] = 0; npos[tid] = 0; plast[tid] = 0; }
    __syncthreads();

    for (int t = 0; t < CHAR_NUM; ++t) {
        // ---- stage enc_t (coalesced: tid == column) + prefetch enc_{t+1} ----
        for (int m = 0; m < BATCH; ++m)
            E[m*ENCW + tid] = enc[((size_t)m * CHAR_NUM + t) * ENCW + tid];
        if ((t + 1) < CHAR_NUM && (tid & 31) == 0) {   // one lane per 128B line
            for (int m = 0; m < BATCH; ++m)
                __builtin_prefetch(&enc[((size_t)m * CHAR_NUM + (t + 1)) * ENCW + tid], 0, 0);
        }
        __syncthreads();

        // ---- logits[16x48] = tanh([H | enc_t] @ W_lin^T + b_lin), waves 0..2 ----
        // Padded cols 33..47 compute garbage (clamped row); they are never read.
        if (wv < 3) {
            int col  = wv * 16 + l15;
            int colc = col < LABELS ? col : (LABELS - 1);   // clamp OOB rows
            v8f acc;
            float bias = b_lin[colc];
            #pragma unroll
            for (int r = 0; r < 8; ++r) acc[r] = bias;
            const float* brow = W_lin + (size_t)colc * KLIN;
            const float* arowH = H + l15 * RNN_H;
            const float* arowE = E + l15 * ENCW;
            for (int k = 0; k < RNN_H; k += 4) {
                int kk = k + koff;
                acc = wmma4(ld2(arowH + kk), ld2(brow + kk), acc);
            }
            for (int k = 0; k < ENCW; k += 4) {
                int kk = k + koff;
                acc = wmma4(ld2(arowE + kk), ld2(brow + RNN_H + kk), acc);
            }
            #pragma unroll
            for (int r = 0; r < 8; ++r) {           // (M=r+8*half, N=col)
                int m = r + 8 * half;
                LG[m*48 + col] = tanhf(acc[r]);
            }
        }
        __syncthreads();

        // ---- per-sequence control: argmax + scalar state (16 lanes) ----
        if (tid < BATCH) {
            int m = tid;
            float best = -INFINITY; int aid = 0;
            for (int l = 0; l < LABELS; ++l) {
                float v = LG[m*48 + l];
                if (v > best) { best = v; aid = l; }   // first max, as jnp.argmax
            }
            int valid = (t < char_sizes[m]) ? 1 : 0;
            int iss   = label_is_sep[aid] > 0;
            int sep   = iss & valid;
            int app   = (!iss) & valid;
            int wl    = wlen[m];
            INVD[m]   = 1.f / (float)(wl > 1 ? wl : 1);     // denom uses OLD wlen
            int npos_n = sep ? npos[m] + 1 : npos[m];
            int nwo    = nw[m];
            int nw_n   = sep ? nwo + 1 : (app ? (nwo > 1 ? nwo : 1) : nwo);
            lpec[m]  = (sep && npos_n >= 2) ? 1 : 0;
            plo[m]   = plast[m];                              // OLD plast
            plast[m] = sep ? label_pos_id[aid] : plast[m];
            wlen[m]  = sep ? 1 : (app ? wl + 1 : wl);
            npos[m]  = npos_n; nw[m] = nw_n;
            dolstm[m] = (sep && nw_n >= 2) ? 1 : 0;
            sepf[m] = sep; validf[m] = valid;
        }
        __syncthreads();

        // ---- vector state updates + masked output write ----
        for (int i = tid; i < BATCH * POS_EMBW; i += NTHREADS) {
            int m = i >> 7;
            if (lpec[m]) LPE[i] = posEmb[plo[m] * POS_EMBW + (i & 127)];
        }
        for (int m = 0; m < BATCH; ++m) {
            int i = m*ENCW + tid;
            float s_old = S[i];
            AVG[i] = (s_old - W0[i]) * INVD[m];   // (prefix[t]-prefix[t-wl])/denom
            if (sepf[m]) W0[i] = s_old;           // new word starts at t -> prefix[t]
            S[i] = s_old + E[i];                  // prefix[t+1]
        }
        for (int i = tid; i < BATCH * LABELS; i += NTHREADS) {
            int m = i / LABELS, l = i - m * LABELS;
            out[((size_t)m * CHAR_NUM + t) * LABELS + l] = validf[m] ? LG[m*48 + l] : 0.f;
        }
        __syncthreads();

        // ---- X[16x384] = tanh([LPE | AVG] @ W_cmb^T + b_cmb), one N-tile/wave ----
        {
            int j = wv * 16 + l15;
            v8f acc;
            float bias = b_cmb[j];
            #pragma unroll
            for (int r = 0; r < 8; ++r) acc[r] = bias;
            const float* brow = W_cmb + (size_t)j * KCMB;
            const float* arowP = LPE + l15 * POS_EMBW;
            const float* arowA = AVG + l15 * ENCW;
            for (int k = 0; k < POS_EMBW; k += 4) {
                int kk = k + koff;
                acc = wmma4(ld2(arowP + kk), ld2(brow + kk), acc);
            }
            for (int k = 0; k < ENCW; k += 4) {
                int kk = k + koff;
                acc = wmma4(ld2(arowA + kk), ld2(brow + POS_EMBW + kk), acc);
            }
            #pragma unroll
            for (int r = 0; r < 8; ++r) {
                int m = r + 8 * half;
                X[m*HIDSZ + j] = tanhf(acc[r]);
            }
        }
        __syncthreads();

        // ---- LSTM gates: G = X@W_ih^T + H@W_hh^T + b, fused nonlinearity ----
        {
            int j = wv * 16 + l15;
            v8f ai, af, ag, ao;
            float bi = b_ih[j]             + b_hh[j];
            float bf = b_ih[RNN_H + j]     + b_hh[RNN_H + j];
            float bg = b_ih[2*RNN_H + j]   + b_hh[2*RNN_H + j];
            float bo = b_ih[3*RNN_H + j]   + b_hh[3*RNN_H + j];
            #pragma unroll
            for (int r = 0; r < 8; ++r) { ai[r]=bi; af[r]=bf; ag[r]=bg; ao[r]=bo; }
            const float* wi0 = W_ih + (size_t)j * HIDSZ;
            const float* wi1 = W_ih + (size_t)(RNN_H   + j) * HIDSZ;
            const float* wi2 = W_ih + (size_t)(2*RNN_H + j) * HIDSZ;
            const float* wi3 = W_ih + (size_t)(3*RNN_H + j) * HIDSZ;
            const float* wh0 = W_hh + (size_t)j * RNN_H;
            const float* wh1 = W_hh + (size_t)(RNN_H   + j) * RNN_H;
            const float* wh2 = W_hh + (size_t)(2*RNN_H + j) * RNN_H;
            const float* wh3 = W_hh + (size_t)(3*RNN_H + j) * RNN_H;
            const float* arowX = X + l15 * HIDSZ;
            const float* arowH = H + l15 * RNN_H;
            for (int k = 0; k < HIDSZ; k += 4) {
                int kk = k + koff;
                v2f ax = ld2(arowX + kk);
                v2f ah = ld2(arowH + kk);
                ai = wmma4(ax, ld2(wi0 + kk), ai);
                af = wmma4(ax, ld2(wi1 + kk), af);
                ag = wmma4(ax, ld2(wi2 + kk), ag);
                ao = wmma4(ax, ld2(wi3 + kk), ao);
                ai = wmma4(ah, ld2(wh0 + kk), ai);
                af = wmma4(ah, ld2(wh1 + kk), af);
                ag = wmma4(ah, ld2(wh2 + kk), ag);
                ao = wmma4(ah, ld2(wh3 + kk), ao);
            }
            float hn[8], cn[8];
            #pragma unroll
            for (int r = 0; r < 8; ++r) {
                int m  = r + 8 * half;
                float ho = H[m*RNN_H + j];
                float co = C[m*RNN_H + j];
                float c2 = sigmoidf_(af[r]) * co + sigmoidf_(ai[r]) * tanhf(ag[r]);
                float h2 = sigmoidf_(ao[r]) * tanhf(c2);
                int dl = dolstm[m];
                hn[r] = dl ? h2 : ho;
                cn[r] = dl ? c2 : co;
            }
            __syncthreads();   // all reads of H/C done before anyone writes
            #pragma unroll
            for (int r = 0; r < 8; ++r) {
                int m = r + 8 * half;
                H[m*RNN_H + j] = hn[r];
                C[m*RNN_H + j] = cn[r];
            }
        }
        __syncthreads();
    } // t
}

extern "C" __global__ void logsoftmax_kernel(float* __restrict__ out, int rows) {
    int r = blockIdx.x * blockDim.x + threadIdx.x;
    if (r >= rows) return;
    float* p = out + (size_t)r * LABELS;
    float mx = p[0];
    for (int l = 1; l < LABELS; ++l) mx = fmaxf(mx, p[l]);
    float s = 0.f;
    for (int l = 0; l < LABELS; ++l) s += expf(p[l] - mx);
    float lse = mx + logf(s);
    for (int l = 0; l < LABELS; ++l) p[l] -= lse;
}

extern "C" void kernel_launch(void* const* d_in, const int* in_sizes, int n_in,
                              void* d_out, int out_size, void* d_ws, size_t ws_size,
                              hipStream_t stream) {
    const float* enc  = (const float*)d_in[0];
    const int*   cs   = (const int*)  d_in[1];
    const int*   lis  = (const int*)  d_in[2];
    const int*   lpi  = (const int*)  d_in[3];
    const float* pe   = (const float*)d_in[4];
    const float* wih  = (const float*)d_in[5];
    const float* whh  = (const float*)d_in[6];
    const float* bih  = (const float*)d_in[7];
    const float* bhh  = (const float*)d_in[8];
    const float* wlin = (const float*)d_in[9];
    const float* blin = (const float*)d_in[10];
    const float* wcmb = (const float*)d_in[11];
    const float* bcmb = (const float*)d_in[12];
    float* out = (float*)d_out;

    size_t smemFloats = (size_t)3 * BATCH * RNN_H    // H, C, X
                      + (size_t)4 * BATCH * ENCW     // E, S, W0, AVG
                      + (size_t)BATCH * POS_EMBW     // LPE
                      + (size_t)BATCH * 48           // LG
                      + 16;                          // INVD
    size_t smemBytes = smemFloats * sizeof(float) + 9 * 16 * sizeof(int);

    hipLaunchKernelGGL(decoder_seq_kernel, dim3(1), dim3(NTHREADS), smemBytes, stream,
                       enc, cs, lis, lpi, pe, wih, whh, bih, bhh,
                       wlin, blin, wcmb, bcmb, out);

    int rows = BATCH * CHAR_NUM;
    hipLaunchKernelGGL(logsoftmax_kernel, dim3((rows + 255) / 256), dim3(256), 0, stream,
                       out, rows);
}